// SGC_7361573945708
// MI455X (gfx1250) — compile-verified
//
#include <hip/hip_runtime.h>
#include <hip/hip_bf16.h>

// ---- problem constants (fixed by reference file) ----
#define N_NODES    100000
#define N_EDGES    1600000
#define IN_F       64
#define OUT_F      40
#define OUT_F_PAD  48          // 3 x 16 WMMA N-tiles
#define K_HOPS     3
#define N_CLUST    10000

typedef __attribute__((ext_vector_type(2))) float v2f;
typedef __attribute__((ext_vector_type(8))) float v8f;

// ---------------- degree: count edges per target row ----------------
__global__ __launch_bounds__(256)
void deg_kernel(const int* __restrict__ row, unsigned* __restrict__ deg) {
    int e = blockIdx.x * 256 + threadIdx.x;
    if (e >= N_EDGES) return;
    atomicAdd(&deg[row[e]], 1u);
}

// dinv[i] = rsqrt(deg_edges[i] + 1)   (+1 = self loop; always >= 1)
__global__ __launch_bounds__(256)
void dinv_kernel(const unsigned* __restrict__ deg, float* __restrict__ dinv) {
    int i = blockIdx.x * 256 + threadIdx.x;
    if (i >= N_NODES) return;
    dinv[i] = rsqrtf((float)(deg[i] + 1u));
}

// ---------------- pad W -> [48][64] and b -> [48] ----------------
__global__ __launch_bounds__(256)
void pad_kernel(const float* __restrict__ W, const float* __restrict__ b,
                float* __restrict__ Wp, float* __restrict__ bp) {
    int t = blockIdx.x * 256 + threadIdx.x;
    if (t < OUT_F_PAD * IN_F) {
        int o = t / IN_F, i = t % IN_F;
        Wp[t] = (o < OUT_F) ? W[o * IN_F + i] : 0.0f;
    } else if (t < OUT_F_PAD * IN_F + OUT_F_PAD) {
        int o = t - OUT_F_PAD * IN_F;
        bp[o] = (o < OUT_F) ? b[o] : 0.0f;
    }
}

// ---------------- one propagation hop (edges + self loops) ----------------
// thread = (edge-or-selfloop, 4-feature chunk); 16 chunks cover IN_F=64
__global__ __launch_bounds__(256)
void hop_kernel(const float* __restrict__ xin, float* __restrict__ xout,
                const int* __restrict__ row, const int* __restrict__ col,
                const float* __restrict__ dinv) {
    int tid = blockIdx.x * 256 + threadIdx.x;
    const int total = (N_EDGES + N_NODES) * 16;
    if (tid >= total) return;
    int e  = tid >> 4;
    int f0 = (tid & 15) * 4;
    int r, c;
    if (e < N_EDGES) { r = row[e]; c = col[e]; }
    else             { r = c = e - N_EDGES; }
    float nrm = dinv[r] * dinv[c];
    const float4 v = *(const float4*)(xin + (size_t)c * IN_F + f0);
    float* dst = xout + (size_t)r * IN_F + f0;
    atomicAdd(dst + 0, v.x * nrm);
    atomicAdd(dst + 1, v.y * nrm);
    atomicAdd(dst + 2, v.z * nrm);
    atomicAdd(dst + 3, v.w * nrm);
}

// ---------------- cluster accumulation ----------------
__global__ __launch_bounds__(256)
void cluster_acc_kernel(const float* __restrict__ x, const int* __restrict__ cidx,
                        float* __restrict__ xcsum, unsigned* __restrict__ cnt) {
    int tid = blockIdx.x * 256 + threadIdx.x;
    if (tid >= N_NODES * 16) return;
    int node = tid >> 4;
    int f0   = (tid & 15) * 4;
    int cl   = cidx[node];
    const float4 v = *(const float4*)(x + (size_t)node * IN_F + f0);
    float* dst = xcsum + (size_t)cl * IN_F + f0;
    atomicAdd(dst + 0, v.x);
    atomicAdd(dst + 1, v.y);
    atomicAdd(dst + 2, v.z);
    atomicAdd(dst + 3, v.w);
    if (f0 == 0) atomicAdd(&cnt[cl], 1u);
}

__global__ __launch_bounds__(256)
void cluster_mean_kernel(const float* __restrict__ xcsum, const unsigned* __restrict__ cnt,
                         float* __restrict__ xc) {
    int tid = blockIdx.x * 256 + threadIdx.x;
    if (tid >= N_CLUST * IN_F) return;
    int c = tid / IN_F;
    xc[tid] = xcsum[tid] / fmaxf((float)cnt[c], 1.0f);
}

// ---------------- dense GEMM via fp32 WMMA: h = xc @ W^T + b ----------------
// One wave per 16x16 output tile.  M tiles: 625 (10000/16), N tiles: 3 (48/16).
// K=64 consumed as 16 chained V_WMMA_F32_16X16X4_F32.
// A (16x4 f32) layout: lanes 0-15 -> K=0,1 ; lanes 16-31 -> K=2,3 (ISA 7.12.2).
// B (4x16) mirrored; C/D: VGPR v holds rows v (lanes 0-15) and v+8 (lanes 16-31).
__global__ __launch_bounds__(128)
void gemm_wmma_kernel(const float* __restrict__ xc,   // [N_CLUST][64]
                      const float* __restrict__ Wp,   // [48][64]  (W^T tiles, padded)
                      const float* __restrict__ bp,   // [48]
                      float* __restrict__ h) {        // [N_CLUST][40]
    const int NT = OUT_F_PAD / 16;                    // 3
    const int totalWaves = (N_CLUST / 16) * NT;       // 1875
    int wave = blockIdx.x * (blockDim.x >> 5) + (threadIdx.x >> 5);
    if (wave >= totalWaves) return;                   // wave-uniform; EXEC stays full
    int mt = wave / NT, nt = wave % NT;
    int lane = threadIdx.x & 31;
    int half = lane >> 4;        // 0: K pair {0,1}, 1: K pair {2,3}
    int l16  = lane & 15;
    int m0 = mt * 16, n0 = nt * 16;

    const float* arow = xc + (size_t)(m0 + l16) * IN_F + 2 * half;
    const float* brow = Wp + (size_t)(n0 + l16) * IN_F + 2 * half;

    v8f c = {};
#pragma unroll
    for (int k0 = 0; k0 < IN_F; k0 += 4) {
        v2f a, b;
        a.x = arow[k0]; a.y = arow[k0 + 1];
        b.x = brow[k0]; b.y = brow[k0 + 1];
        // (neg_a, A, neg_b, B, c_mod, C, reuse_a, reuse_b)
        c = __builtin_amdgcn_wmma_f32_16x16x4_f32(false, a, false, b,
                                                  (short)0, c, false, false);
    }

    int n = n0 + l16;
    float bias = bp[n];
#pragma unroll
    for (int v = 0; v < 8; ++v) {
        int m = m0 + v + half * 8;
        if (n < OUT_F) h[(size_t)m * OUT_F + n] = c[v] + bias;
    }
}

// ---------------- gather rows back to nodes ----------------
__global__ __launch_bounds__(256)
void gather_kernel(const float* __restrict__ h, const int* __restrict__ cidx,
                   float* __restrict__ out) {
    int tid = blockIdx.x * 256 + threadIdx.x;
    if (tid >= N_NODES * (OUT_F / 4)) return;
    int node = tid / (OUT_F / 4);
    int f0   = (tid % (OUT_F / 4)) * 4;
    int cl   = cidx[node];
    *(float4*)(out + (size_t)node * OUT_F + f0) =
        *(const float4*)(h + (size_t)cl * OUT_F + f0);
}

static inline int blocks(long long work, int bs) { return (int)((work + bs - 1) / bs); }

extern "C" void kernel_launch(void* const* d_in, const int* in_sizes, int n_in,
                              void* d_out, int out_size, void* d_ws, size_t ws_size,
                              hipStream_t stream) {
    const float* x    = (const float*)d_in[0];           // [N][64]
    const int*   ei   = (const int*)d_in[1];             // [2][E]
    const int*   cidx = (const int*)d_in[2];             // [N]
    // d_in[3] = num_clusters scalar (compile-time constant here)
    const float* W    = (const float*)d_in[4];           // [40][64]
    const float* b    = (const float*)d_in[5];           // [40]
    float* out = (float*)d_out;

    const int* row = ei;
    const int* col = ei + N_EDGES;

    // ---- carve workspace (256B aligned) ----
    char* ws = (char*)d_ws;
    size_t off = 0;
    auto carve = [&](size_t bytes) { char* p = ws + off;
                                     off = (off + bytes + 255) & ~(size_t)255; return p; };
    unsigned* deg   = (unsigned*)carve((size_t)N_NODES * 4);
    float*    dinv  = (float*)   carve((size_t)N_NODES * 4);
    float*    bufA  = (float*)   carve((size_t)N_NODES * IN_F * 4);
    float*    bufB  = (float*)   carve((size_t)N_NODES * IN_F * 4);
    float*    xcsum = (float*)   carve((size_t)N_CLUST * IN_F * 4);
    unsigned* cnt   = (unsigned*)carve((size_t)N_CLUST * 4);
    float*    xc    = (float*)   carve((size_t)N_CLUST * IN_F * 4);
    float*    Wp    = (float*)   carve((size_t)OUT_F_PAD * IN_F * 4);
    float*    bp    = (float*)   carve((size_t)OUT_F_PAD * 4);
    float*    h     = (float*)   carve((size_t)N_CLUST * OUT_F * 4);
    (void)ws_size;

    // ---- normalization setup ----
    hipMemsetAsync(deg, 0, (size_t)N_NODES * 4, stream);
    deg_kernel<<<blocks(N_EDGES, 256), 256, 0, stream>>>(row, deg);
    dinv_kernel<<<blocks(N_NODES, 256), 256, 0, stream>>>(deg, dinv);
    pad_kernel<<<blocks(OUT_F_PAD * IN_F + OUT_F_PAD, 256), 256, 0, stream>>>(W, b, Wp, bp);

    // ---- K=3 propagation hops (ping-pong, dst zeroed each hop) ----
    const long long hopWork = (long long)(N_EDGES + N_NODES) * 16;
    const float* src = x;
    float* dst = bufA;
    float* other = bufB;
    for (int k = 0; k < K_HOPS; ++k) {
        hipMemsetAsync(dst, 0, (size_t)N_NODES * IN_F * 4, stream);
        hop_kernel<<<blocks(hopWork, 256), 256, 0, stream>>>(src, dst, row, col, dinv);
        src = dst;
        float* t = dst == bufA ? bufB : bufA; (void)other;
        dst = t;
    }
    const float* xfinal = src;   // result of 3rd hop

    // ---- cluster mean ----
    hipMemsetAsync(xcsum, 0, (size_t)N_CLUST * IN_F * 4, stream);
    hipMemsetAsync(cnt,   0, (size_t)N_CLUST * 4, stream);
    cluster_acc_kernel<<<blocks((long long)N_NODES * 16, 256), 256, 0, stream>>>(
        xfinal, cidx, xcsum, cnt);
    cluster_mean_kernel<<<blocks((long long)N_CLUST * IN_F, 256), 256, 0, stream>>>(
        xcsum, cnt, xc);

    // ---- dense GEMM on compressed rows: fp32 WMMA ----
    const int totalWaves = (N_CLUST / 16) * (OUT_F_PAD / 16);   // 1875
    gemm_wmma_kernel<<<blocks(totalWaves, 4), 128, 0, stream>>>(xc, Wp, bp, h);

    // ---- reconstruct per-node output ----
    gather_kernel<<<blocks((long long)N_NODES * (OUT_F / 4), 256), 256, 0, stream>>>(
        h, cidx, out);

    (void)in_sizes; (void)n_in; (void)out_size;
}